// MultiHeadSelfAttention_89807766160028
// MI455X (gfx1250) — compile-verified
//
#include <hip/hip_runtime.h>

typedef __attribute__((ext_vector_type(2))) float v2f;
typedef __attribute__((ext_vector_type(4))) float v4f;
typedef __attribute__((ext_vector_type(8))) float v8f;

#define DIM_      1024
#define NHEADS_   16
#define HDIM_     64
#define B_        2
#define T_        2048

#define BM 128
#define BN 64
#define BK 32
#define PADA 4   // As row stride 36 floats = 144 B: 16B-aligned rows, conflict-free frag reads
#define PADB 2   // Bs row stride 34 floats: 8B-aligned even k, conflict-free frag reads

// ---------------------------------------------------------------------------
// fp32 WMMA GEMM:  C = op(A) * op(B) (+ bias / scale / causal mask)
//   A_MODE 0: A[m*lda+k]        1: gather merged-heads from [B,H,T,HDIM]
//   B_MODE 0: B[k*ldb+n]        1: B[n*ldb+k] (transposed read, for Q K^T)
//   S_MODE 0: plain (+bias)     1: split-heads store (+bias)
//          2: scores: *inv_scale, causal mask (-inf where n > m)
// 256 threads = 8 wave32; block tile 128x64; wave tile 32x32 (2x2 wmma accs).
// Global->LDS staging goes through registers with next-K-tile prefetch
// overlapped with compute; fragment loads are software-pipelined.
// ---------------------------------------------------------------------------
template <int A_MODE, int B_MODE, int S_MODE>
__global__ __launch_bounds__(256) void gemm_wmma_f32(
    const float* __restrict__ A, const float* __restrict__ Bm,
    const float* __restrict__ bias, float* __restrict__ C,
    int K, int lda, int ldb, int ldc,
    size_t strideA, size_t strideB, size_t strideC, float inv_scale)
{
  A  += (size_t)blockIdx.z * strideA;
  Bm += (size_t)blockIdx.z * strideB;
  C  += (size_t)blockIdx.z * strideC;

  const int tid   = threadIdx.x;
  const int lane  = tid & 31;
  const int wave  = tid >> 5;
  const int waveM = wave >> 1;  // 0..3
  const int waveN = wave & 1;   // 0..1
  const int l15   = lane & 15;
  const int kbOff = (lane < 16) ? 0 : 2;  // frag K split across lane halves

  const int m0 = blockIdx.y * BM;
  const int n0 = blockIdx.x * BN;

  if constexpr (S_MODE == 2) {
    // Tile entirely above the causal diagonal: vector-fill -inf, skip GEMM.
    if (n0 >= m0 + BM) {
      const v4f ninf = {-__builtin_inff(), -__builtin_inff(),
                        -__builtin_inff(), -__builtin_inff()};
      #pragma unroll
      for (int i = 0; i < (BM * BN) / (256 * 4); ++i) {
        int idx4 = tid + i * 256;
        int r = idx4 >> 4, c4 = idx4 & 15;
        *(v4f*)(C + (unsigned)((m0 + r) * ldc + n0 + c4 * 4)) = ninf;
      }
      return;
    }
  }

  __shared__ float As[BM][BK + PADA];  // k-major, frag pair contiguous
  __shared__ float Bs[BN][BK + PADB];  // k-major (n rows), frag pair contiguous

  // ---- register-staged global fetch (all 16B global_load_b128) ----
  auto fetchA = [&](int k0, v4f ra[4]) {
    #pragma unroll
    for (int i = 0; i < 4; ++i) {
      int idx4 = tid + i * 256;            // over [BM][BK/4]
      int r = idx4 >> 3, c4 = idx4 & 7;
      int gm = m0 + r, gk = k0 + c4 * 4;
      if constexpr (A_MODE == 0) {
        ra[i] = *(const v4f*)(A + (unsigned)(gm * lda + gk));
      } else {  // merged-heads gather: 4 consecutive hd never cross a head
        int bb = gm >> 11, t = gm & (T_ - 1);
        int head = gk >> 6, hd = gk & (HDIM_ - 1);
        ra[i] = *(const v4f*)(A + (unsigned)((((bb * NHEADS_) + head) * T_ + t) * HDIM_ + hd));
      }
    }
  };
  auto fetchB = [&](int k0, v4f rb[2]) {
    #pragma unroll
    for (int i = 0; i < 2; ++i) {
      int idx4 = tid + i * 256;
      if constexpr (B_MODE == 0) {         // over [BK][BN/4], n-contiguous
        int k = idx4 >> 4, n4 = idx4 & 15;
        rb[i] = *(const v4f*)(Bm + (unsigned)((k0 + k) * ldb + n0 + n4 * 4));
      } else {                             // over [BN][BK/4], k-contiguous
        int n = idx4 >> 3, k4 = idx4 & 7;
        rb[i] = *(const v4f*)(Bm + (unsigned)((n0 + n) * ldb + k0 + k4 * 4));
      }
    }
  };
  auto stageA = [&](v4f ra[4]) {
    #pragma unroll
    for (int i = 0; i < 4; ++i) {
      int idx4 = tid + i * 256;
      int r = idx4 >> 3, c4 = idx4 & 7;
      *(v4f*)&As[r][c4 * 4] = ra[i];       // 144B row stride -> b128 store
    }
  };
  auto stageB = [&](v4f rb[2]) {
    #pragma unroll
    for (int i = 0; i < 2; ++i) {
      int idx4 = tid + i * 256;
      if constexpr (B_MODE == 0) {         // transpose into k-major LDS
        int k = idx4 >> 4, n4 = idx4 & 15;
        #pragma unroll
        for (int j = 0; j < 4; ++j) Bs[n4 * 4 + j][k] = rb[i][j];
      } else {
        int n = idx4 >> 3, k4 = idx4 & 7;
        #pragma unroll
        for (int j = 0; j < 4; ++j) Bs[n][k4 * 4 + j] = rb[i][j];
      }
    }
  };
  auto loadFrag = [&](int kk, v2f a[2], v2f b[2]) {
    const int kb = kk + kbOff;
    #pragma unroll
    for (int t2 = 0; t2 < 2; ++t2) {
      int ar = waveM * 32 + t2 * 16 + l15;
      a[t2].x = As[ar][kb];
      a[t2].y = As[ar][kb + 1];
      int bc = waveN * 32 + t2 * 16 + l15;
      b[t2].x = Bs[bc][kb];
      b[t2].y = Bs[bc][kb + 1];
    }
  };

  v8f acc[2][2] = {};
  v4f ra[4], rb[2];
  fetchA(0, ra);
  fetchB(0, rb);

  for (int k0 = 0; k0 < K; k0 += BK) {
    stageA(ra);
    stageB(rb);
    __syncthreads();

    // prefetch next K-tile while computing this one
    if (k0 + BK < K) {
      fetchA(k0 + BK, ra);
      fetchB(k0 + BK, rb);
    }

    v2f a_c[2], b_c[2];
    loadFrag(0, a_c, b_c);
    #pragma unroll
    for (int kk = 0; kk < BK; kk += 4) {
      v2f a_n[2], b_n[2];
      if (kk + 4 < BK) loadFrag(kk + 4, a_n, b_n);  // pipeline next frags
      #pragma unroll
      for (int ti = 0; ti < 2; ++ti)
        #pragma unroll
        for (int tj = 0; tj < 2; ++tj)
          acc[ti][tj] = __builtin_amdgcn_wmma_f32_16x16x4_f32(
              false, a_c[ti], false, b_c[tj], (short)0, acc[ti][tj], false, false);
      if (kk + 4 < BK) {
        #pragma unroll
        for (int t2 = 0; t2 < 2; ++t2) { a_c[t2] = a_n[t2]; b_c[t2] = b_n[t2]; }
      }
    }
    __syncthreads();
  }

  // ---- store: C/D layout VGPR i -> row i (lanes 0-15) / row 8+i (lanes 16-31)
  #pragma unroll
  for (int ti = 0; ti < 2; ++ti) {
    #pragma unroll
    for (int tj = 0; tj < 2; ++tj) {
      const int col   = n0 + waveN * 32 + tj * 16 + l15;
      const int rbase = m0 + waveM * 32 + ti * 16 + ((lane < 16) ? 0 : 8);
      #pragma unroll
      for (int i = 0; i < 8; ++i) {
        const int row = rbase + i;
        float v = acc[ti][tj][i];
        if constexpr (S_MODE == 0) {
          if (bias) v += bias[col];
          C[(unsigned)(row * ldc + col)] = v;
        } else if constexpr (S_MODE == 1) {
          if (bias) v += bias[col];
          int bb = row >> 11, t = row & (T_ - 1);
          int head = col >> 6, hd = col & (HDIM_ - 1);
          C[(unsigned)((((bb * NHEADS_) + head) * T_ + t) * HDIM_ + hd)] = v;
        } else {
          v = (col <= row) ? v * inv_scale : -__builtin_inff();
          C[(unsigned)(row * ldc + col)] = v;
        }
      }
    }
  }
}

// ---------------------------------------------------------------------------
// In-place row softmax over attn. One 256-thread block per 2048-wide row;
// masked entries are -inf -> exp 0. float4 I/O, wave32 shuffle + LDS reduce.
// ---------------------------------------------------------------------------
__global__ __launch_bounds__(256) void softmax_rows(float* __restrict__ attn)
{
  v4f* p4 = (v4f*)(attn + (size_t)blockIdx.x * T_);
  const int tid = threadIdx.x;
  const int lane = tid & 31, wid = tid >> 5;
  __shared__ float red[8];

  v4f x[2];
  float m = -__builtin_inff();
  #pragma unroll
  for (int i = 0; i < 2; ++i) {
    x[i] = p4[tid + i * 256];
    #pragma unroll
    for (int j = 0; j < 4; ++j) m = fmaxf(m, x[i][j]);
  }
  #pragma unroll
  for (int off = 16; off > 0; off >>= 1) m = fmaxf(m, __shfl_xor(m, off, 32));
  if (lane == 0) red[wid] = m;
  __syncthreads();
  float bm = -__builtin_inff();
  #pragma unroll
  for (int j = 0; j < 8; ++j) bm = fmaxf(bm, red[j]);
  __syncthreads();

  float s = 0.f;
  #pragma unroll
  for (int i = 0; i < 2; ++i)
    #pragma unroll
    for (int j = 0; j < 4; ++j) {
      x[i][j] = expf(x[i][j] - bm);  // exp(-inf - finite) == 0
      s += x[i][j];
    }
  #pragma unroll
  for (int off = 16; off > 0; off >>= 1) s += __shfl_xor(s, off, 32);
  if (lane == 0) red[wid] = s;
  __syncthreads();
  float tot = 0.f;
  #pragma unroll
  for (int j = 0; j < 8; ++j) tot += red[j];
  const float inv = 1.0f / tot;
  #pragma unroll
  for (int i = 0; i < 2; ++i) {
    v4f y = x[i];
    #pragma unroll
    for (int j = 0; j < 4; ++j) y[j] *= inv;
    p4[tid + i * 256] = y;
  }
}

// ---------------------------------------------------------------------------
extern "C" void kernel_launch(void* const* d_in, const int* in_sizes, int n_in,
                              void* d_out, int out_size, void* d_ws, size_t ws_size,
                              hipStream_t stream)
{
  const float* x  = (const float*)d_in[0];
  const float* Wq = (const float*)d_in[1];
  const float* bq = (const float*)d_in[2];
  const float* Wk = (const float*)d_in[3];
  const float* bk = (const float*)d_in[4];
  const float* Wv = (const float*)d_in[5];
  const float* bv = (const float*)d_in[6];
  const float* Wo = (const float*)d_in[7];
  const float* bo = (const float*)d_in[8];

  float* out  = (float*)d_out;                 // [B,T,DIM]
  float* attn = out + (size_t)B_ * T_ * DIM_;  // [B,H,T,T]

  // Workspace: Q,K,V each [B,H,T,HDIM] = 16 MB; O aliases Q (dead after scores).
  const size_t qn = (size_t)B_ * NHEADS_ * T_ * HDIM_;
  float* Q  = (float*)d_ws;
  float* Km = Q + qn;
  float* V  = Q + 2 * qn;
  float* O  = Q;

  const size_t sQK = (size_t)T_ * HDIM_;  // per-(b,h) Q/K/V stride
  const size_t sAT = (size_t)T_ * T_;     // per-(b,h) attn stride

  dim3 blk(256);
  dim3 gproj(DIM_ / BN, (B_ * T_) / BM, 1);     // 16 x 32
  dim3 gsc(T_ / BN, T_ / BM, B_ * NHEADS_);     // 32 x 16 x 32
  dim3 gpv(HDIM_ / BN, T_ / BM, B_ * NHEADS_);  // 1 x 16 x 32

  // 1) QKV projections with fused split-heads store
  gemm_wmma_f32<0, 0, 1><<<gproj, blk, 0, stream>>>(
      x, Wq, bq, Q, DIM_, DIM_, DIM_, DIM_, 0, 0, 0, 1.0f);
  gemm_wmma_f32<0, 0, 1><<<gproj, blk, 0, stream>>>(
      x, Wk, bk, Km, DIM_, DIM_, DIM_, DIM_, 0, 0, 0, 1.0f);
  gemm_wmma_f32<0, 0, 1><<<gproj, blk, 0, stream>>>(
      x, Wv, bv, V, DIM_, DIM_, DIM_, DIM_, 0, 0, 0, 1.0f);

  // 2) scores = Q K^T / sqrt(64), causal mask at store (raw into attn region)
  gemm_wmma_f32<0, 1, 2><<<gsc, blk, 0, stream>>>(
      Q, Km, nullptr, attn, HDIM_, HDIM_, HDIM_, T_, sQK, sQK, sAT, 0.125f);

  // 3) row softmax in place
  softmax_rows<<<dim3(B_ * NHEADS_ * T_), blk, 0, stream>>>(attn);

  // 4) O = attn @ V   (per (b,h))
  gemm_wmma_f32<0, 0, 0><<<gpv, blk, 0, stream>>>(
      attn, V, nullptr, O, T_, T_, HDIM_, HDIM_, sAT, sQK, sQK, 1.0f);

  // 5) out = merge_heads(O) @ Wo + bo
  gemm_wmma_f32<1, 0, 0><<<gproj, blk, 0, stream>>>(
      O, Wo, bo, out, DIM_, DIM_, DIM_, DIM_, 0, 0, 0, 1.0f);
}